// Qwen3MoeFusedSparseMoeBlock_90117003804876
// MI455X (gfx1250) — compile-verified
//
#include <hip/hip_runtime.h>
#include <hip/hip_bf16.h>

// ---------------- problem constants ----------------
#define NEXP   8
#define TOPK   2
#define HID    2048
#define INTER  1408
#define MTOK   2048
#define CAP    2048          // max rows per expert (top-2 distinct => <= MTOK)

#define ASTR      36         // A LDS row stride (floats): 16B aligned, conflict-free b64 frags
#define BSTR2_GU  144        // B LDS pair-row stride in float2 units (128 cols + pad)
#define BSTR2_DN  80         // B LDS pair-row stride in float2 units (64 cols + pad)

typedef float v2f __attribute__((ext_vector_type(2)));
typedef float v8f __attribute__((ext_vector_type(8)));

// D = A(16x4 f32) * B(4x16 f32) + C(16x16 f32), wave32
static __device__ __forceinline__ v8f wmma4(v2f a, v2f b, v8f c) {
    return __builtin_amdgcn_wmma_f32_16x16x4_f32(false, a, false, b, (short)0, c,
                                                 false, false);
}

static __device__ __forceinline__ float silu(float x) {
    return x / (1.0f + __expf(-x));
}

// generic LDS pointer -> raw LDS byte offset (ISA: LDS generic addr low 32 bits)
static __device__ __forceinline__ unsigned lds_off(const void* p) {
    return (unsigned)(unsigned long long)p;
}

// async copy 16B / 4B global -> LDS (per lane), tracked by ASYNCcnt
static __device__ __forceinline__ void async_ld_b128(unsigned loff, const float* g) {
    asm volatile("global_load_async_to_lds_b128 %0, %1, off"
                 :: "v"(loff), "v"(g)
                 : "memory");
}
static __device__ __forceinline__ void async_ld_b32(unsigned loff, const float* g) {
    asm volatile("global_load_async_to_lds_b32 %0, %1, off"
                 :: "v"(loff), "v"(g)
                 : "memory");
}

// ---------------- kernel 0: zero output + counters ----------------
__global__ void moe_zero_kernel(float* __restrict__ out, int* __restrict__ counts) {
    size_t i = (size_t)blockIdx.x * blockDim.x + threadIdx.x;
    float4 z = {0.f, 0.f, 0.f, 0.f};
    ((float4*)out)[i] = z;                       // MTOK*HID/4 elements exactly
    if (blockIdx.x == 0 && threadIdx.x < NEXP) counts[threadIdx.x] = 0;
}

// ---------------- kernel 1: router (one wave per token) ----------------
__global__ void moe_router_kernel(const float* __restrict__ hs,
                                  const float* __restrict__ gw,
                                  float* __restrict__ topk_w,
                                  int* __restrict__ counts,
                                  int* __restrict__ expert_rows) {
    const int wave  = threadIdx.x >> 5;
    const int lane  = threadIdx.x & 31;
    const int token = blockIdx.x * 8 + wave;
    const float* __restrict__ hrow = hs + (size_t)token * HID;

    float acc[NEXP];
#pragma unroll
    for (int e = 0; e < NEXP; ++e) acc[e] = 0.f;

    for (int i = lane; i < HID; i += 32) {
        float x = hrow[i];
#pragma unroll
        for (int e = 0; e < NEXP; ++e) acc[e] += x * gw[e * HID + i];
    }
#pragma unroll
    for (int e = 0; e < NEXP; ++e) {
#pragma unroll
        for (int m = 16; m >= 1; m >>= 1) acc[e] += __shfl_xor(acc[e], m, 32);
    }

    if (lane == 0) {
        // stable top-2 on logits (softmax is monotone)
        int   i0 = 0;
        float l0 = acc[0];
#pragma unroll
        for (int e = 1; e < NEXP; ++e)
            if (acc[e] > l0) { l0 = acc[e]; i0 = e; }
        int   i1 = (i0 == 0) ? 1 : 0;
        float l1 = acc[i1];
#pragma unroll
        for (int e = 0; e < NEXP; ++e)
            if (e != i0 && acc[e] > l1) { l1 = acc[e]; i1 = e; }

        // renormalized top-2 softmax weights
        float ew = __expf(l1 - l0);
        float w0 = 1.0f / (1.0f + ew);
        float w1 = ew   / (1.0f + ew);

        topk_w[token * 2 + 0] = w0;
        topk_w[token * 2 + 1] = w1;

        int p0 = atomicAdd(&counts[i0], 1);
        expert_rows[i0 * CAP + p0] = token * 2;
        int p1 = atomicAdd(&counts[i1], 1);
        expert_rows[i1 * CAP + p1] = token * 2 + 1;
    }
}

// ---------------- kernel 2: grouped gate_up GEMM + SiLU*mul ----------------
// block: 64 rows x 64 h-cols (= 128 GEMM cols: gate[0..63] + up[64..127])
// async global->LDS double-buffered pipeline; B pair-swizzled for b64 fragments
__global__ __launch_bounds__(256)
void moe_gateup_kernel(const float* __restrict__ hs,
                       const float* __restrict__ gup,
                       const int*   __restrict__ counts,
                       const int*   __restrict__ expert_rows,
                       float*       __restrict__ hbuf) {
    const int e   = blockIdx.z;
    const int rt  = blockIdx.y;
    const int n0  = blockIdx.x * 64;
    const int cnt = counts[e];
    if (rt * 64 >= cnt) return;
    const int* __restrict__ rows_e = expert_rows + e * CAP;

    __shared__ float Al[2][64 * ASTR];
    __shared__ float Bl[2][16 * BSTR2_GU * 2];

    const int tid  = threadIdx.x;
    const int lane = tid & 31;
    const int wv   = tid >> 5;
    const int wm   = wv >> 1;      // 0..3 : row group
    const int wn   = wv & 1;       // 0..1 : 32-wide h-col group
    const int lh   = lane >> 4;    // lane half
    const int lm   = lane & 15;

    const size_t wbase = (size_t)e * HID * (2 * INTER);

    // --- A staging: 2 x b128 per thread (64x32 tile = 512 float4) ---
    const float* gA[2];
    unsigned     lA[2];
#pragma unroll
    for (int i = 0; i < 2; ++i) {
        int f = tid + i * 256;
        int row = f >> 3, col = (f & 7) * 4;
        int r  = rt * 64 + row;
        int cr = (r < cnt) ? r : (cnt - 1);
        int flat = rows_e[cr];
        gA[i] = hs + (size_t)(flat >> 1) * HID + col;
        lA[i] = lds_off(&Al[0][row * ASTR + col]);
    }
    // --- B staging: 16 x b32 per thread (32x128 tile = 4096 floats) ---
    // f = tid + i*256 : col = tid & 127 (const), k = (tid>>7) + 2*i
    const int bc  = tid & 127;               // tile column (const per thread)
    const int bk0 = tid >> 7;                // 0 or 1 : k parity/base
    const int bgc = (bc < 64) ? (n0 + bc) : (INTER + n0 + (bc - 64));
    const float* gBp = gup + wbase + (size_t)bk0 * (2 * INTER) + bgc;
    const unsigned lBp = lds_off(&Bl[0][bc * 2 + (bk0 & 1)]);  // pair-row 0 slot

    const unsigned ABUF = 64 * ASTR * 4;         // bytes per A buffer
    const unsigned BBUF = 16 * BSTR2_GU * 2 * 4; // bytes per B buffer
    const unsigned BROW = BSTR2_GU * 2 * 4;      // bytes per LDS pair-row

    v8f aG0 = {}, aG1 = {}, aU0 = {}, aU1 = {};
    const int NT = HID / 32;

    // prologue: tile 0 -> buffer 0
#pragma unroll
    for (int i = 0; i < 2; ++i) async_ld_b128(lA[i], gA[i]);
#pragma unroll
    for (int i = 0; i < 16; ++i)
        async_ld_b32(lBp + (unsigned)i * BROW, gBp + (size_t)(2 * i) * (2 * INTER));

    for (int kt = 0; kt < NT; ++kt) {
        const int cb = kt & 1;
        if (kt + 1 < NT) {
            const unsigned ao = ((kt + 1) & 1) ? ABUF : 0;
            const unsigned bo = ((kt + 1) & 1) ? BBUF : 0;
            const float* gBt = gBp + (size_t)(kt + 1) * 32 * (2 * INTER);
#pragma unroll
            for (int i = 0; i < 2; ++i)
                async_ld_b128(lA[i] + ao, gA[i] + (kt + 1) * 32);
#pragma unroll
            for (int i = 0; i < 16; ++i)
                async_ld_b32(lBp + bo + (unsigned)i * BROW,
                             gBt + (size_t)(2 * i) * (2 * INTER));
            asm volatile("s_wait_asynccnt 0x12" ::: "memory");  // tile kt landed
        } else {
            asm volatile("s_wait_asynccnt 0x0" ::: "memory");
        }
        __syncthreads();

        const float* Ab = &Al[cb][0];
        const float* Bb = &Bl[cb][0];
#pragma unroll
        for (int kk = 0; kk < 8; ++kk) {
            const int k4 = kk * 4;
            v2f a = *(const v2f*)(Ab + (wm * 16 + lm) * ASTR + k4 + 2 * lh);
            const float* bp = Bb + ((k4 >> 1) + lh) * BSTR2_GU * 2;
            v2f bg0 = *(const v2f*)(bp + (wn * 32 + lm) * 2);
            v2f bg1 = *(const v2f*)(bp + (wn * 32 + 16 + lm) * 2);
            v2f bu0 = *(const v2f*)(bp + (64 + wn * 32 + lm) * 2);
            v2f bu1 = *(const v2f*)(bp + (64 + wn * 32 + 16 + lm) * 2);
            aG0 = wmma4(a, bg0, aG0);
            aG1 = wmma4(a, bg1, aG1);
            aU0 = wmma4(a, bu0, aU0);
            aU1 = wmma4(a, bu1, aU1);
        }
        __syncthreads();
    }

    // epilogue: h = silu(gate) * up
#pragma unroll
    for (int j = 0; j < 8; ++j) {
        int row = rt * 64 + wm * 16 + j + 8 * lh;
        if (row < cnt) {
            int flat = rows_e[row];
            float* hr = hbuf + (size_t)flat * INTER + n0 + wn * 32;
            hr[lm]      = silu(aG0[j]) * aU0[j];
            hr[16 + lm] = silu(aG1[j]) * aU1[j];
        }
    }
}

// ---------------- kernel 3: grouped down GEMM + weighted combine ----------------
// block: 64 rows x 64 out-cols, async double-buffered, B pair-swizzled
__global__ __launch_bounds__(256)
void moe_down_kernel(const float* __restrict__ hbuf,
                     const float* __restrict__ dw,
                     const int*   __restrict__ counts,
                     const int*   __restrict__ expert_rows,
                     const float* __restrict__ topk_w,
                     float*       __restrict__ out) {
    const int e   = blockIdx.z;
    const int rt  = blockIdx.y;
    const int n0  = blockIdx.x * 64;
    const int cnt = counts[e];
    if (rt * 64 >= cnt) return;
    const int* __restrict__ rows_e = expert_rows + e * CAP;

    __shared__ float Al[2][64 * ASTR];
    __shared__ float Bl[2][16 * BSTR2_DN * 2];

    const int tid  = threadIdx.x;
    const int lane = tid & 31;
    const int wv   = tid >> 5;
    const int wm   = wv >> 1;
    const int wn   = wv & 1;
    const int lh   = lane >> 4;
    const int lm   = lane & 15;

    const size_t wbase = (size_t)e * INTER * HID;

    // --- A staging: 2 x b128 per thread ---
    const float* gA[2];
    unsigned     lA[2];
#pragma unroll
    for (int i = 0; i < 2; ++i) {
        int f = tid + i * 256;
        int row = f >> 3, col = (f & 7) * 4;
        int r  = rt * 64 + row;
        int cr = (r < cnt) ? r : (cnt - 1);
        int flat = rows_e[cr];
        gA[i] = hbuf + (size_t)flat * INTER + col;
        lA[i] = lds_off(&Al[0][row * ASTR + col]);
    }
    // --- B staging: 8 x b32 per thread (32x64 tile = 2048 floats) ---
    // f = tid + i*256 : col = tid & 63 (const), k = (tid>>6) + 4*i
    const int bc  = tid & 63;
    const int bk0 = tid >> 6;                 // 0..3
    const float* gBp = dw + wbase + (size_t)bk0 * HID + n0 + bc;
    const unsigned lBp =
        lds_off(&Bl[0][((bk0 >> 1) * BSTR2_DN + bc) * 2 + (bk0 & 1)]);

    const unsigned ABUF  = 64 * ASTR * 4;
    const unsigned BBUF  = 16 * BSTR2_DN * 2 * 4;
    const unsigned BROW2 = 2 * BSTR2_DN * 2 * 4;  // bytes per 2 pair-rows (k += 4)

    v8f a0 = {}, a1 = {};
    const int NT = INTER / 32;

#pragma unroll
    for (int i = 0; i < 2; ++i) async_ld_b128(lA[i], gA[i]);
#pragma unroll
    for (int i = 0; i < 8; ++i)
        async_ld_b32(lBp + (unsigned)i * BROW2, gBp + (size_t)(4 * i) * HID);

    for (int kt = 0; kt < NT; ++kt) {
        const int cb = kt & 1;
        if (kt + 1 < NT) {
            const unsigned ao = ((kt + 1) & 1) ? ABUF : 0;
            const unsigned bo = ((kt + 1) & 1) ? BBUF : 0;
            const float* gBt = gBp + (size_t)(kt + 1) * 32 * HID;
#pragma unroll
            for (int i = 0; i < 2; ++i)
                async_ld_b128(lA[i] + ao, gA[i] + (kt + 1) * 32);
#pragma unroll
            for (int i = 0; i < 8; ++i)
                async_ld_b32(lBp + bo + (unsigned)i * BROW2,
                             gBt + (size_t)(4 * i) * HID);
            asm volatile("s_wait_asynccnt 0xa" ::: "memory");
        } else {
            asm volatile("s_wait_asynccnt 0x0" ::: "memory");
        }
        __syncthreads();

        const float* Ab = &Al[cb][0];
        const float* Bb = &Bl[cb][0];
#pragma unroll
        for (int kk = 0; kk < 8; ++kk) {
            const int k4 = kk * 4;
            v2f a = *(const v2f*)(Ab + (wm * 16 + lm) * ASTR + k4 + 2 * lh);
            const float* bp = Bb + ((k4 >> 1) + lh) * BSTR2_DN * 2;
            v2f b0 = *(const v2f*)(bp + (wn * 32 + lm) * 2);
            v2f b1 = *(const v2f*)(bp + (wn * 32 + 16 + lm) * 2);
            a0 = wmma4(a, b0, a0);
            a1 = wmma4(a, b1, a1);
        }
        __syncthreads();
    }

    // epilogue: out[token] += w * val  (exactly two commutative adds per element)
#pragma unroll
    for (int j = 0; j < 8; ++j) {
        int row = rt * 64 + wm * 16 + j + 8 * lh;
        if (row < cnt) {
            int flat  = rows_e[row];
            int token = flat >> 1;
            float w   = topk_w[flat];
            float* orow = out + (size_t)token * HID + n0 + wn * 32;
            atomicAdd(orow + lm,      w * a0[j]);
            atomicAdd(orow + 16 + lm, w * a1[j]);
        }
    }
}

// ---------------- host launch ----------------
extern "C" void kernel_launch(void* const* d_in, const int* in_sizes, int n_in,
                              void* d_out, int out_size, void* d_ws, size_t ws_size,
                              hipStream_t stream) {
    const float* hs  = (const float*)d_in[0];   // [M, H]
    const float* gw  = (const float*)d_in[1];   // [E, H]
    const float* gup = (const float*)d_in[2];   // [E, H, 2*INTER]
    const float* dw  = (const float*)d_in[3];   // [E, INTER, H]
    float* out = (float*)d_out;                 // [M, H]

    char* ws = (char*)d_ws;
    int*   counts      = (int*)(ws + 0);                   // 8 ints
    float* topk_w      = (float*)(ws + 1024);              // M*2 floats
    int*   expert_rows = (int*)(ws + 1024 + 16384);        // 8*CAP ints
    float* hbuf        = (float*)(ws + 131072);            // (M*TOPK) * INTER floats

    moe_zero_kernel<<<(MTOK * HID / 4) / 256, 256, 0, stream>>>(out, counts);
    moe_router_kernel<<<MTOK / 8, 256, 0, stream>>>(hs, gw, topk_w, counts, expert_rows);
    moe_gateup_kernel<<<dim3(INTER / 64, CAP / 64, NEXP), 256, 0, stream>>>(
        hs, gup, counts, expert_rows, hbuf);
    moe_down_kernel<<<dim3(HID / 64, CAP / 64, NEXP), 256, 0, stream>>>(
        hbuf, dw, counts, expert_rows, topk_w, out);
}